// MoE_11424613007529
// MI455X (gfx1250) — compile-verified
//
#include <hip/hip_runtime.h>
#include <hip/hip_bf16.h>

typedef __attribute__((ext_vector_type(2))) float v2f;
typedef __attribute__((ext_vector_type(8))) float v8f;

#define D_DIM 1024
#define H_DIM 4096
#define E_NUM 8
#define NTOK  8192      // B*S = 4*2048
#define MT    32        // tokens per block tile
#define HC    128       // H chunk staged through LDS
#define XS    1028      // x_lds row stride (mod 64 == 4 -> conflict-free, 8B aligned)
#define HS    132       // h_lds row stride (mod 64 == 4)

__device__ __forceinline__ float gelu_exact(float v) {
    return 0.5f * v * (1.0f + erff(v * 0.70710678118654752f));
}

__device__ __forceinline__ v8f wmma_f32(v2f a, v2f b, v8f c) {
    return __builtin_amdgcn_wmma_f32_16x16x4_f32(false, a, false, b, (short)0, c,
                                                 false, false);
}

// ---------------------------------------------------------------------------
// Kernel 0: zero output + per-expert counters (deterministic per launch)
// ---------------------------------------------------------------------------
__global__ void zero_out_kernel(float* __restrict__ out, size_t n,
                                int* __restrict__ counts) {
    size_t i = (size_t)blockIdx.x * blockDim.x + threadIdx.x;
    size_t stride = (size_t)gridDim.x * blockDim.x;
    for (; i < n; i += stride) out[i] = 0.0f;
    if (blockIdx.x == 0 && threadIdx.x < E_NUM) counts[threadIdx.x] = 0;
}

// ---------------------------------------------------------------------------
// Kernel 1: gating -> top-2 routing, compacted per-expert lists
// ---------------------------------------------------------------------------
__global__ __launch_bounds__(256) void gate_kernel(
    const float* __restrict__ x, const float* __restrict__ gw,
    const float* __restrict__ gb, const float* __restrict__ temp,
    int* __restrict__ counts, int* __restrict__ etok, float* __restrict__ ewt) {
    const int wv = threadIdx.x >> 5;
    const int ln = threadIdx.x & 31;
    const int token = blockIdx.x * 8 + wv;   // 8 waves/block, 1 token/wave
    if (token >= NTOK) return;

    const float* xr = x + (size_t)token * D_DIM;
    float s[E_NUM];
#pragma unroll
    for (int e = 0; e < E_NUM; ++e) s[e] = 0.0f;

    for (int k = ln; k < D_DIM; k += 32) {
        float xv = xr[k];
        const float* g = gw + (size_t)k * E_NUM;
#pragma unroll
        for (int e = 0; e < E_NUM; ++e) s[e] += xv * g[e];
    }
#pragma unroll
    for (int off = 16; off > 0; off >>= 1) {
#pragma unroll
        for (int e = 0; e < E_NUM; ++e) s[e] += __shfl_xor(s[e], off, 32);
    }
    if (ln == 0) {
        float t = temp[0];
        float g[E_NUM];
#pragma unroll
        for (int e = 0; e < E_NUM; ++e) g[e] = (s[e] + gb[e]) / t;
        int i0 = 0;
#pragma unroll
        for (int e = 1; e < E_NUM; ++e) if (g[e] > g[i0]) i0 = e;
        int i1 = (i0 == 0) ? 1 : 0;
#pragma unroll
        for (int e = 0; e < E_NUM; ++e)
            if (e != i0 && g[e] > g[i1]) i1 = e;
        // softmax over {g[i0] >= g[i1]}
        float e1 = __expf(g[i1] - g[i0]);
        float w0 = 1.0f / (1.0f + e1);
        float w1 = e1 / (1.0f + e1);
        int s0 = atomicAdd(&counts[i0], 1);
        etok[i0 * NTOK + s0] = token; ewt[i0 * NTOK + s0] = w0;
        int s1 = atomicAdd(&counts[i1], 1);
        etok[i1 * NTOK + s1] = token; ewt[i1 * NTOK + s1] = w1;
    }
}

// ---------------------------------------------------------------------------
// Kernel 2: fused routed FFN  out += w * (gelu(x@W1+b1) @ W2 + b2)
//   grid.x = token tiles (MT each), grid.y = expert; 8 waves/block
//   per wave: GEMM1 -> one 16-col h slice; GEMM2 -> 128 output cols
// ---------------------------------------------------------------------------
__global__ __launch_bounds__(256, 1) void moe_ffn_kernel(
    const float* __restrict__ x, const float* __restrict__ W1,
    const float* __restrict__ b1, const float* __restrict__ W2,
    const float* __restrict__ b2, const int* __restrict__ counts,
    const int* __restrict__ etok, const float* __restrict__ ewt,
    float* __restrict__ out) {
    __shared__ float x_lds[MT][XS];
    __shared__ float h_lds[MT][HS];
    __shared__ int   tok_s[MT];
    __shared__ float wt_s[MT];

    const int e = blockIdx.y;
    const int cnt = counts[e];
    const int row0 = blockIdx.x * MT;
    if (row0 >= cnt) return;

    const int tid = threadIdx.x;
    const int wv = tid >> 5;
    const int ln = tid & 31;

    if (tid < MT) {
        int r = row0 + tid;
        int rc = (r < cnt) ? r : (cnt - 1);
        tok_s[tid] = etok[e * NTOK + rc];
        wt_s[tid] = (r < cnt) ? ewt[e * NTOK + rc] : 0.0f;  // padded rows: weight 0
    }
    __syncthreads();

    // stage x tile: each wave loads 4 rows, float4-coalesced
#pragma unroll
    for (int rr = 0; rr < 4; ++rr) {
        int m = wv * 4 + rr;
        const float* src = x + (size_t)tok_s[m] * D_DIM;
        for (int k = ln * 4; k < D_DIM; k += 32 * 4) {
            float4 v = *(const float4*)(src + k);
            *(float4*)&x_lds[m][k] = v;
        }
    }
    __syncthreads();

    const float* W1e = W1 + (size_t)e * D_DIM * H_DIM;
    const float* W2e = W2 + (size_t)e * H_DIM * D_DIM;

    v8f acc[16];
#pragma unroll
    for (int i = 0; i < 16; ++i) acc[i] = {};

    const int mfrag = ln & 15;        // A: row (lanes 0-15 / 16-31)
    const int koff = (ln >> 4) * 2;   // A/B: K sub-offset per half-wave

    for (int hc = 0; hc < H_DIM; hc += HC) {
        // ---- phase 1: h[MT][HC] = gelu(x @ W1[:, hc:hc+HC] + b1) ----
        const int hcol = hc + wv * 16 + mfrag;
        v8f c0 = {}, c1 = {};
        for (int kb = 0; kb < D_DIM; kb += 4) {
            v2f a0 = *(const v2f*)&x_lds[mfrag][kb + koff];
            v2f a1 = *(const v2f*)&x_lds[16 + mfrag][kb + koff];
            int k0 = kb + koff;
            v2f b;
            b.x = W1e[(size_t)k0 * H_DIM + hcol];
            b.y = W1e[(size_t)(k0 + 1) * H_DIM + hcol];
            c0 = wmma_f32(a0, b, c0);
            c1 = wmma_f32(a1, b, c1);
        }
        float bias1 = b1[e * H_DIM + hcol];
#pragma unroll
        for (int i = 0; i < 8; ++i) {
            int m = i + 8 * (ln >> 4);
            h_lds[m][wv * 16 + mfrag] = gelu_exact(c0[i] + bias1);
            h_lds[16 + m][wv * 16 + mfrag] = gelu_exact(c1[i] + bias1);
        }
        __syncthreads();
        // ---- phase 2: acc += h_chunk @ W2[hc:hc+HC, wv*128 : wv*128+128] ----
        for (int kb = 0; kb < HC; kb += 4) {
            v2f a0 = *(const v2f*)&h_lds[mfrag][kb + koff];
            v2f a1 = *(const v2f*)&h_lds[16 + mfrag][kb + koff];
            int krow = hc + kb + koff;
#pragma unroll
            for (int t = 0; t < 8; ++t) {
                int nc = wv * 128 + t * 16 + mfrag;
                v2f b;
                b.x = W2e[(size_t)krow * D_DIM + nc];
                b.y = W2e[(size_t)(krow + 1) * D_DIM + nc];
                acc[2 * t] = wmma_f32(a0, b, acc[2 * t]);
                acc[2 * t + 1] = wmma_f32(a1, b, acc[2 * t + 1]);
            }
        }
        __syncthreads();
    }

    // ---- epilogue: + b2, scale by routing weight, atomic accumulate ----
#pragma unroll
    for (int t = 0; t < 8; ++t) {
        int nc = wv * 128 + t * 16 + mfrag;
        float bias2 = b2[e * D_DIM + nc];
#pragma unroll
        for (int i = 0; i < 8; ++i) {
            int m0 = i + 8 * (ln >> 4);
            float v0 = (acc[2 * t][i] + bias2) * wt_s[m0];
            atomicAdd(&out[(size_t)tok_s[m0] * D_DIM + nc], v0);
            int m1 = 16 + m0;
            float v1 = (acc[2 * t + 1][i] + bias2) * wt_s[m1];
            atomicAdd(&out[(size_t)tok_s[m1] * D_DIM + nc], v1);
        }
    }
}

// ---------------------------------------------------------------------------
extern "C" void kernel_launch(void* const* d_in, const int* in_sizes, int n_in,
                              void* d_out, int out_size, void* d_ws, size_t ws_size,
                              hipStream_t stream) {
    (void)in_sizes; (void)n_in; (void)ws_size;
    const float* x      = (const float*)d_in[0];
    const float* W1     = (const float*)d_in[1];
    const float* b1     = (const float*)d_in[2];
    const float* W2     = (const float*)d_in[3];
    const float* b2     = (const float*)d_in[4];
    const float* gate_w = (const float*)d_in[5];
    const float* gate_b = (const float*)d_in[6];
    const float* temp   = (const float*)d_in[7];
    float* out = (float*)d_out;

    // workspace layout: [counts: 8 ints | pad to 256B | etok: E*NTOK ints | ewt: E*NTOK floats]
    char* ws = (char*)d_ws;
    int*   counts = (int*)ws;
    int*   etok   = (int*)(ws + 256);
    float* ewt    = (float*)(ws + 256 + (size_t)E_NUM * NTOK * sizeof(int));

    zero_out_kernel<<<512, 256, 0, stream>>>(out, (size_t)out_size, counts);
    gate_kernel<<<NTOK / 8, 256, 0, stream>>>(x, gate_w, gate_b, temp,
                                              counts, etok, ewt);
    dim3 grid(NTOK / MT, E_NUM);
    moe_ffn_kernel<<<grid, 256, 0, stream>>>(x, W1, b1, W2, b2,
                                             counts, etok, ewt, out);
}